// ClusterGCN_56092272885943
// MI455X (gfx1250) — compile-verified
//
#include <hip/hip_runtime.h>

#define NN 50000
#define NE 400000
#define INC 770
#define HID 32
#define ROWTILES (NN / 16)          // 3125 exactly
#define COLGROUPS 13                // ceil(770/64)
#define G2TILES (ROWTILES * COLGROUPS)

typedef float v2f __attribute__((ext_vector_type(2)));
typedef float v8f __attribute__((ext_vector_type(8)));

__device__ __forceinline__ v8f wmma4(v2f a, v2f b, v8f c) {
  // D = A(16x4) * B(4x16) + C(16x16), fp32 WMMA
  return __builtin_amdgcn_wmma_f32_16x16x4_f32(false, a, false, b, (short)0, c,
                                               false, false);
}
__device__ __forceinline__ v2f ld2(const float* p) { return *(const v2f*)p; }

// ---------------- elementwise / graph kernels ----------------
__global__ void zero_f32(float* __restrict__ p, int n) {
  int i = blockIdx.x * blockDim.x + threadIdx.x;
  if (i < n) p[i] = 0.f;
}

__global__ void deg_count(const int* __restrict__ row, const int* __restrict__ col,
                          float* __restrict__ deg, int E) {
  int e = blockIdx.x * blockDim.x + threadIdx.x;
  if (e < E) {
    int r = row[e], c = col[e];
    if (r != c) atomicAdd(&deg[c], 1.0f);
  }
}

__global__ void recip_deg(float* __restrict__ deg, int n) {
  int i = blockIdx.x * blockDim.x + threadIdx.x;
  if (i < n) deg[i] = 1.0f / (deg[i] + 1.0f);
}

// dst[i*32+c] = src[i*srcStride+c] * degInv[i] * 2.0 (self-loop term, diag_lambda=1)
__global__ void init_agg(float* __restrict__ dst, const float* __restrict__ src,
                         int srcStride, const float* __restrict__ degInv, int total) {
  int t = blockIdx.x * blockDim.x + threadIdx.x;
  if (t < total) {
    int i = t >> 5, c = t & 31;
    dst[t] = src[(long)i * srcStride + c] * degInv[i] * 2.0f;
  }
}

// one wave handles one edge (32 channels); row/col/deginv wave-uniform
__global__ void scatter_edges(const int* __restrict__ row, const int* __restrict__ col,
                              const float* __restrict__ src, int srcStride,
                              const float* __restrict__ degInv, float* __restrict__ dst,
                              int E) {
  int t = blockIdx.x * blockDim.x + threadIdx.x;
  int e = t >> 5, c = t & 31;
  if (e < E) {
    int r = row[e], cl = col[e];
    if (r != cl)
      atomicAdd(&dst[(long)cl * HID + c], src[(long)r * srcStride + c] * degInv[cl]);
  }
}

// h = relu(agg1 + b_out1 + r), in place into agg1 buffer; r = y1[:,32:64]
__global__ void relu_add(float* __restrict__ hbuf, const float* __restrict__ y1,
                         const float* __restrict__ bias, int total) {
  int t = blockIdx.x * blockDim.x + threadIdx.x;
  if (t < total) {
    int i = t >> 5, c = t & 31;
    float v = hbuf[t] + bias[c] + y1[(long)i * 64 + 32 + c];
    hbuf[t] = v > 0.f ? v : 0.f;
  }
}

// ---------------- GEMM1: y(N x 64) = x(N x 770) @ [Wout1;Wroot1]^T ----------------
__global__ __launch_bounds__(256) void gemm1(const float* __restrict__ x,
                                             const float* __restrict__ Wout,
                                             const float* __restrict__ Wroot,
                                             float* __restrict__ y, int nRowTiles) {
  int wave = (blockIdx.x * 256 + threadIdx.x) >> 5;
  int lane = threadIdx.x & 31;
  if (wave >= nRowTiles) return;
  int half = lane >> 4, n = lane & 15;
  long m0 = (long)wave * 16;

  const float* arow = x + (m0 + n) * INC;        // A: M = lane%16
  const float* b0r  = Wout  + (long)n * INC;     // B tiles n0 = 0,16,32,48
  const float* b1r  = Wout  + (long)(n + 16) * INC;
  const float* b2r  = Wroot + (long)n * INC;
  const float* b3r  = Wroot + (long)(n + 16) * INC;

  v8f z = {0.f, 0.f, 0.f, 0.f, 0.f, 0.f, 0.f, 0.f};
  v8f acc0 = z, acc1 = z, acc2 = z, acc3 = z;
  int kk0 = 2 * half;

  for (int k0 = 0; k0 < 768; k0 += 4) {
    int kk = k0 + kk0;
    v2f a = ld2(arow + kk);
    acc0 = wmma4(a, ld2(b0r + kk), acc0);
    acc1 = wmma4(a, ld2(b1r + kk), acc1);
    acc2 = wmma4(a, ld2(b2r + kk), acc2);
    acc3 = wmma4(a, ld2(b3r + kk), acc3);
  }
  // K tail: k = 768,769 valid (half 0); half 1 would read 770,771 -> zero pad
  {
    v2f a = {0.f, 0.f}, b0 = {0.f, 0.f}, b1 = {0.f, 0.f}, b2 = {0.f, 0.f}, b3 = {0.f, 0.f};
    if (half == 0) {
      a = ld2(arow + 768);
      b0 = ld2(b0r + 768); b1 = ld2(b1r + 768);
      b2 = ld2(b2r + 768); b3 = ld2(b3r + 768);
    }
    acc0 = wmma4(a, b0, acc0);
    acc1 = wmma4(a, b1, acc1);
    acc2 = wmma4(a, b2, acc2);
    acc3 = wmma4(a, b3, acc3);
  }
#pragma unroll
  for (int v = 0; v < 8; ++v) {
    float* p = y + (m0 + v + 8 * half) * 64 + n;   // D[M=v+8*half][N=n]
    p[0]  = acc0[v];
    p[16] = acc1[v];
    p[32] = acc2[v];
    p[48] = acc3[v];
  }
}

// ---------------- GEMM2: out(N x 770) = [agg2|h](N x 64) @ [Wout2|Wroot2]^T + b ----
__global__ __launch_bounds__(256) void gemm2(const float* __restrict__ agg2,
                                             const float* __restrict__ h,
                                             const float* __restrict__ Wout2,
                                             const float* __restrict__ Wroot2,
                                             const float* __restrict__ bias,
                                             float* __restrict__ out, int nTiles) {
  int wave = (blockIdx.x * 256 + threadIdx.x) >> 5;
  int lane = threadIdx.x & 31;
  if (wave >= nTiles) return;
  int rowTile = wave / COLGROUPS;
  int g = wave - rowTile * COLGROUPS;
  int half = lane >> 4, n = lane & 15;
  long m0 = (long)rowTile * 16;
  int base = g * 64;

  const float* aA = agg2 + (m0 + n) * HID;   // K = 0..31
  const float* aH = h    + (m0 + n) * HID;   // K = 32..63
  v8f z = {0.f, 0.f, 0.f, 0.f, 0.f, 0.f, 0.f, 0.f};
  v8f acc[4] = {z, z, z, z};
  int kk0 = 2 * half;

#pragma unroll
  for (int s = 0; s < 16; ++s) {
    int kk = 4 * s + kk0;
    v2f a = (s < 8) ? ld2(aA + kk) : ld2(aH + kk - 32);
#pragma unroll
    for (int t = 0; t < 4; ++t) {
      int o = base + t * 16 + n;
      v2f b = {0.f, 0.f};
      if (o < INC)
        b = (s < 8) ? ld2(Wout2 + (long)o * HID + kk)
                    : ld2(Wroot2 + (long)o * HID + kk - 32);
      acc[t] = wmma4(a, b, acc[t]);
    }
  }
#pragma unroll
  for (int t = 0; t < 4; ++t) {
    int o = base + t * 16 + n;
    if (o < INC) {
      float bb = bias[o];
#pragma unroll
      for (int v = 0; v < 8; ++v)
        out[(m0 + v + 8 * half) * INC + o] = acc[t][v] + bb;
    }
  }
}

// ---------------- launcher ----------------
extern "C" void kernel_launch(void* const* d_in, const int* in_sizes, int n_in,
                              void* d_out, int out_size, void* d_ws, size_t ws_size,
                              hipStream_t stream) {
  const float* x      = (const float*)d_in[0];
  const int*   ei     = (const int*)d_in[1];
  const float* Wout1  = (const float*)d_in[2];
  const float* bout1  = (const float*)d_in[3];
  const float* Wroot1 = (const float*)d_in[4];
  const float* Wout2  = (const float*)d_in[5];
  const float* bout2  = (const float*)d_in[6];
  const float* Wroot2 = (const float*)d_in[7];
  float* out = (float*)d_out;

  float* fws    = (float*)d_ws;
  float* degInv = fws;                         // N
  float* y1     = fws + NN;                    // N x 64 (u | r); later reused as agg2
  float* aggA   = y1 + (size_t)NN * 64;        // N x 32 (agg1 -> h in place)
  const int* row = ei;
  const int* col = ei + NE;

  const int B = 256;
  // degree + inverse
  zero_f32<<<(NN + B - 1) / B, B, 0, stream>>>(degInv, NN);
  deg_count<<<(NE + B - 1) / B, B, 0, stream>>>(row, col, degInv, NE);
  recip_deg<<<(NN + B - 1) / B, B, 0, stream>>>(degInv, NN);
  // layer 1: project first, then 32-wide aggregation
  gemm1<<<(ROWTILES + 7) / 8, B, 0, stream>>>(x, Wout1, Wroot1, y1, ROWTILES);
  init_agg<<<(NN * 32 + B - 1) / B, B, 0, stream>>>(aggA, y1, 64, degInv, NN * 32);
  scatter_edges<<<(NE * 32 + B - 1) / B, B, 0, stream>>>(row, col, y1, 64, degInv, aggA, NE);
  relu_add<<<(NN * 32 + B - 1) / B, B, 0, stream>>>(aggA, y1, bout1, NN * 32);
  // layer 2: aggregate h, then fused output GEMM
  init_agg<<<(NN * 32 + B - 1) / B, B, 0, stream>>>(y1, aggA, 32, degInv, NN * 32);
  scatter_edges<<<(NE * 32 + B - 1) / B, B, 0, stream>>>(row, col, aggA, 32, degInv, y1, NE);
  gemm2<<<(G2TILES + 7) / 8, B, 0, stream>>>(y1, aggA, Wout2, Wroot2, bout2, out, G2TILES);
}